// AudioRandomLowpassFilter_22711787061985
// MI455X (gfx1250) — compile-verified
//
#include <hip/hip_runtime.h>
#include <math.h>

typedef __attribute__((ext_vector_type(2))) float v2f;
typedef __attribute__((ext_vector_type(8))) float v8f;
typedef __attribute__((ext_vector_type(4))) unsigned int u32x4;
typedef __attribute__((ext_vector_type(8))) int i32x8;
typedef __attribute__((ext_vector_type(4))) int i32x4;

#define BATCH   64
#define TLEN    480000
#define CG      16                 // channels per workgroup
#define TCHUNK  256                // output timesteps per workgroup
#define TAPW    48                 // FIR history window (|h[48]| ~ 1e-15)
#define TW      (TCHUNK + TAPW)    // 304 floats staged per channel row
#define TWP     308                // padded LDS row stride (floats): bank-stride 52 -> conflict-free gathers
#define NTILE   (TCHUNK / 16)      // 16 output tiles per chunk
#define NWAVE   8
#define NTHREADS (NWAVE * 32)

#if defined(__has_builtin)
#if __has_builtin(__builtin_amdgcn_tensor_load_to_lds) && __has_builtin(__builtin_amdgcn_s_wait_tensorcnt)
#define USE_TDM 1
#endif
#endif

// D = A x B + C  with  A (16x4) = input data [channel m x time j],
// B (4x16) = Toeplitz h[n-j] [time j x out-time n],  D = y[channel m, t0+n].
__global__ __launch_bounds__(NTHREADS)
void biquad_fir_wmma(const float* __restrict__ wav, float* __restrict__ out,
                     float b0, float b1, float b2, float a1, float a2)
{
    __shared__ float XL[CG * TWP];   // staged input tile [channel][time], padded rows
    __shared__ float HH[64];         // biquad impulse response taps

    const int tid  = threadIdx.x;
    const int lane = tid & 31;
    const int wave = tid >> 5;
    const int half = lane >> 4;      // 0: lanes 0-15, 1: lanes 16-31
    const int l15  = lane & 15;
    const int cbase = blockIdx.y * CG;
    const long T0   = (long)blockIdx.x * TCHUNK;
    const bool edge = (blockIdx.x == 0);

    // ---- impulse response of the biquad (exact recurrence, 64 taps) ----
    if (tid == 0) {
        float x1 = 0.f, x2 = 0.f, y1 = 0.f, y2 = 0.f;
        for (int k = 0; k < 64; ++k) {
            float x = (k == 0) ? 1.f : 0.f;
            float y = b0 * x + b1 * x1 + b2 * x2 - a1 * y1 - a2 * y2;
            HH[k] = y;
            x2 = x1; x1 = x; y2 = y1; y1 = y;
        }
    }

#ifdef USE_TDM
    // zero-fill the t<0 warm-up region for the first chunk (zero initial filter state)
    if (edge) {
        for (int q = tid; q < CG * TAPW; q += NTHREADS) {
            XL[(q / TAPW) * TWP + (q % TAPW)] = 0.f;
        }
    }
    // wave 0 issues one 1-D TDM row load per channel (scalar branch: tensor ops ignore EXEC)
    if (__builtin_amdgcn_readfirstlane(tid >> 5) == 0) {
        const unsigned lds_base = (unsigned)(unsigned long long)(&XL[0]);
        const unsigned td0 = edge ? (unsigned)TCHUNK : (unsigned)TW;   // row length (dwords)
        const long gt0 = edge ? 0 : (T0 - TAPW);
        for (int r = 0; r < CG; ++r) {
            unsigned long long ga =
                (unsigned long long)(wav + (long)(cbase + r) * TLEN + gt0);
            unsigned lb = lds_base + (unsigned)(r * TWP + (edge ? TAPW : 0)) * 4u;
            u32x4 g0;
            g0[0] = 1u;                                   // count = 1 valid descriptor
            g0[1] = lb;                                   // lds_addr (bytes)
            g0[2] = (unsigned)ga;                         // global_addr[31:0]
            g0[3] = (unsigned)(ga >> 32) | (2u << 30);    // global_addr[56:32] | type=2
            i32x8 g1;
            g1[0] = (int)(2u << 16);                      // data_size = 4 bytes
            g1[1] = (int)(td0 << 16);                     // tensor_dim0[15:0]
            g1[2] = (int)(1u << 16);                      // tensor_dim0 hi=0 | tensor_dim1=1
            g1[3] = (int)(td0 << 16);                     // tensor_dim1 hi=0 | tile_dim0
            g1[4] = 1;                                    // tile_dim1 = 1, tile_dim2 = 0
            g1[5] = (int)td0;                             // tensor_dim0_stride[31:0]
            g1[6] = 0;                                    // stride0 hi | stride1 lo
            g1[7] = 0;                                    // stride1 hi
            i32x4 z4 = {0, 0, 0, 0};
            i32x8 z8 = {0, 0, 0, 0, 0, 0, 0, 0};
            __builtin_amdgcn_tensor_load_to_lds(g0, g1, z4, z4, z8, 0);
        }
        __builtin_amdgcn_s_wait_tensorcnt(0);
    }
#else
    // fallback staging: vectorized global loads + LDS stores
    const int NV = CG * (TW / 4);
    for (int q = tid; q < NV; q += NTHREADS) {
        int r  = q / (TW / 4);
        int tq = (q % (TW / 4)) * 4;
        long tg = T0 - TAPW + tq;
        float4 val = make_float4(0.f, 0.f, 0.f, 0.f);
        if (tg >= 0)
            val = *(const float4*)(wav + (long)(cbase + r) * TLEN + tg);
        *(float4*)(&XL[r * TWP + tq]) = val;
    }
#endif
    __syncthreads();

    // ---- per-wave constant Toeplitz B matrices for the 16 K-steps ----
    // step s covers j = -TAPW + 4*s + K.  B VGPR v holds K = v + 2*half.
    // B[K][n] = h[n - j] = HH[l15 + TAPW - 4*s - K]
    v2f Bm[16];
#pragma unroll
    for (int s = 0; s < 16; ++s) {
        int d0 = l15 + TAPW - 4 * s - 2 * half;
        int d1 = d0 - 1;
        Bm[s].x = (d0 >= 0 && d0 < 64) ? HH[d0] : 0.f;
        Bm[s].y = (d1 >= 0 && d1 < 64) ? HH[d1] : 0.f;
    }

    // ---- each wave computes two tiles straight-line (dual accumulators for ILP) ----
    const int tA = wave;
    const int tB = wave + NWAVE;
    v8f acc0 = {0.f, 0.f, 0.f, 0.f, 0.f, 0.f, 0.f, 0.f};
    v8f acc1 = {0.f, 0.f, 0.f, 0.f, 0.f, 0.f, 0.f, 0.f};
    const float* rowp = &XL[l15 * TWP + 2 * half];
#pragma unroll
    for (int s = 0; s < 16; ++s) {
        // A: lane holds M=l15 (channel), K = 2*half, 2*half+1 (consecutive times)
        const float* p0 = rowp + tA * 16 + 4 * s;
        const float* p1 = rowp + tB * 16 + 4 * s;
        v2f a0, a1v;
        a0.x  = p0[0]; a0.y  = p0[1];
        a1v.x = p1[0]; a1v.y = p1[1];
        acc0 = __builtin_amdgcn_wmma_f32_16x16x4_f32(
            false, a0, false, Bm[s], (short)0, acc0, false, false);
        acc1 = __builtin_amdgcn_wmma_f32_16x16x4_f32(
            false, a1v, false, Bm[s], (short)0, acc1, false, false);
    }

    // D layout: lane gives n = l15 (time), VGPR v gives channel = v + 8*half
    const long t0A = T0 + (long)tA * 16;
    const long t0B = T0 + (long)tB * 16;
#pragma unroll
    for (int v = 0; v < 8; ++v) {
        long crow = (long)(cbase + v + 8 * half) * TLEN;
        out[crow + t0A + l15] = acc0[v];
        out[crow + t0B + l15] = acc1[v];
    }
}

extern "C" void kernel_launch(void* const* d_in, const int* in_sizes, int n_in,
                              void* d_out, int out_size, void* d_ws, size_t ws_size,
                              hipStream_t stream) {
    const float* wav = (const float*)d_in[0];
    float* out = (float*)d_out;

    // torchaudio lowpass_biquad coefficients (host-side double precision)
    const double SR = 24000.0, CUTOFF = 8000.0, Q = 0.707;
    double w0    = 2.0 * M_PI * CUTOFF / SR;
    double alpha = sin(w0) / (2.0 * Q);
    double cw    = cos(w0);
    double a0    = 1.0 + alpha;
    float b0 = (float)(((1.0 - cw) / 2.0) / a0);
    float b1 = (float)((1.0 - cw) / a0);
    float b2 = b0;
    float a1 = (float)((-2.0 * cw) / a0);
    float a2 = (float)((1.0 - alpha) / a0);

    dim3 grid(TLEN / TCHUNK, BATCH / CG);   // 1875 x 4 workgroups
    biquad_fir_wmma<<<grid, NTHREADS, 0, stream>>>(wav, out, b0, b1, b2, a1, a2);
}